// EnhancedCompositeSeq2SeqLoss_8194797601171
// MI455X (gfx1250) — compile-verified
//
#include <hip/hip_runtime.h>
#include <math.h>

// ---------------- problem constants (from reference) ----------------
#define BB   16
#define TT   128
#define VV   32000
#define HH   768
#define PP   384
#define NBOW 64
#define EPSV 0.05f
#define TAUV 0.07f
#define W_CE  1.0f
#define W_AL  0.5f
#define W_BOW 0.2f
#define W_DIV 0.1f
#define W_VAR 0.05f

typedef __attribute__((ext_vector_type(2))) float v2f;
typedef __attribute__((ext_vector_type(8))) float v8f;

// =====================================================================
// Kernel 1: label-smoothed CE, one block per token, single streaming
// pass over V=32000 logits (online logsumexp + running sum of logits).
// =====================================================================
__global__ __launch_bounds__(256)
void ce_kernel(const float* __restrict__ logits, const int* __restrict__ labels,
               float* __restrict__ tok_out, float* __restrict__ vf_out)
{
    const int tok = blockIdx.x;                    // 0 .. B*T-1
    const float* __restrict__ row = logits + (size_t)tok * VV;
    const int tid = threadIdx.x;

    float m = -INFINITY, s = 0.0f, ssum = 0.0f;
    const float4* row4 = reinterpret_cast<const float4*>(row);
    for (int i = tid; i < VV / 4; i += 256) {
        float4 x4 = row4[i];
        float xs[4] = {x4.x, x4.y, x4.z, x4.w};
#pragma unroll
        for (int k = 0; k < 4; ++k) {
            float x = xs[k];
            ssum += x;
            float nm = fmaxf(m, x);
            s = s * __expf(m - nm) + __expf(x - nm);
            m = nm;
        }
    }

    __shared__ float sm[256], ss[256], st[256];
    sm[tid] = m; ss[tid] = s; st[tid] = ssum;
    __syncthreads();
    for (int off = 128; off > 0; off >>= 1) {
        if (tid < off) {
            float m1 = sm[tid], s1 = ss[tid];
            float m2 = sm[tid + off], s2 = ss[tid + off];
            float nm = fmaxf(m1, m2);
            sm[tid] = nm;
            ss[tid] = s1 * __expf(m1 - nm) + s2 * __expf(m2 - nm);
            st[tid] += st[tid + off];
        }
        __syncthreads();
    }

    if (tid == 0) {
        int   lab   = labels[tok];
        bool  valid = (lab != 0) && (lab != -100);
        int   lc    = min(max(lab, 0), VV - 1);
        float xl    = row[lc];
        float lse   = sm[0] + __logf(ss[0]);
        float lp_lab = xl - lse;
        float sum_lp = st[0] - (float)VV * lse;
        float tl = -((1.0f - EPSV) * lp_lab + (EPSV / (float)VV) * sum_lp);
        tok_out[tok] = valid ? tl : 0.0f;
        vf_out[tok]  = valid ? 1.0f : 0.0f;
    }
}

// =====================================================================
// Kernel 2: masked mean-pool of decoder_hidden over T. One block per b.
// =====================================================================
__global__ __launch_bounds__(256)
void pool_kernel(const float* __restrict__ dh, const int* __restrict__ mask,
                 float* __restrict__ pooled)
{
    const int b = blockIdx.x;
    const int tid = threadIdx.x;

    float cnt = 0.0f;
    for (int t = 0; t < TT; ++t) cnt += (float)mask[b * TT + t];
    cnt = fmaxf(cnt, 1.0f);

    for (int h = tid; h < HH; h += 256) {
        float acc = 0.0f;
        for (int t = 0; t < TT; ++t)
            acc += dh[((size_t)b * TT + t) * HH + h] * (float)mask[b * TT + t];
        pooled[b * HH + h] = acc / cnt;
    }
}

// =====================================================================
// Kernel 3: projection head  z = l2norm( gelu(LN(x)@W1+b1) @ W2 + b2 )
// One block, 8 waves (wave32).  GEMMs use V_WMMA_F32_16X16X4_F32:
// M=16 rows, N tiles of 16 (24 tiles -> 3 per wave), K stepped by 4.
// A frag (16x4 f32): lanes 0-15 hold K{0,1}, lanes 16-31 hold K{2,3}.
// B frag (4x16 f32): VGPR0 = rows K0/K2, VGPR1 = rows K1/K3, N=lane&15.
// C/D (16x16 f32): VGPR j -> M=j (lanes 0-15) / M=j+8 (lanes 16-31).
// =====================================================================
__global__ __launch_bounds__(256)
void proj_wmma_kernel(const float* __restrict__ xin,          // 16 x H
                      const float* __restrict__ g,  const float* __restrict__ bn,
                      const float* __restrict__ W1, const float* __restrict__ b1, // H x P
                      const float* __restrict__ W2, const float* __restrict__ b2, // P x P
                      float* __restrict__ zout)                // 16 x P (l2-normalized)
{
    __shared__ float buf[16 * HH];   // 48 KB: xn (16xH), later h (16xP) + y (16xP)
    const int tid  = threadIdx.x;
    const int wid  = tid >> 5;
    const int lane = tid & 31;
    const int arow = lane & 15;
    const int kh   = (lane >> 4) * 2;   // 0 for lanes 0-15, 2 for lanes 16-31

    v8f czero;
#pragma unroll
    for (int k = 0; k < 8; ++k) czero[k] = 0.0f;

    // ---- LayerNorm rows: wave w handles rows w and w+8 ----
    for (int r = wid; r < 16; r += 8) {
        const float* xr = xin + r * HH;
        float sum = 0.0f;
        for (int c = lane; c < HH; c += 32) sum += xr[c];
#pragma unroll
        for (int o = 16; o > 0; o >>= 1) sum += __shfl_xor(sum, o, 32);
        float mean = sum / (float)HH;
        float var = 0.0f;
        for (int c = lane; c < HH; c += 32) { float d = xr[c] - mean; var += d * d; }
#pragma unroll
        for (int o = 16; o > 0; o >>= 1) var += __shfl_xor(var, o, 32);
        float rstd = rsqrtf(var / (float)HH + 1e-5f);
        for (int c = lane; c < HH; c += 32)
            buf[r * HH + c] = (xr[c] - mean) * rstd * g[c] + bn[c];
    }
    __syncthreads();

    // ---- GEMM1: (16xH)@(HxP), results kept in registers ----
    v8f acc[3];
#pragma unroll
    for (int t = 0; t < 3; ++t) {
        const int n0 = (wid + t * 8) * 16;
        v8f c = czero;
        for (int k = 0; k < HH; k += 4) {
            v2f a, bfr;
            a[0]   = buf[arow * HH + k + kh];
            a[1]   = buf[arow * HH + k + kh + 1];
            bfr[0] = W1[(size_t)(k + kh    ) * PP + n0 + arow];
            bfr[1] = W1[(size_t)(k + kh + 1) * PP + n0 + arow];
            c = __builtin_amdgcn_wmma_f32_16x16x4_f32(false, a, false, bfr,
                                                      (short)0, c, false, false);
        }
        acc[t] = c;
    }
    __syncthreads();   // all reads of xn done -> safe to overwrite buf

    // ---- h = gelu(acc + b1) into buf[0 .. 16*P) ----
#pragma unroll
    for (int t = 0; t < 3; ++t) {
        const int col = (wid + t * 8) * 16 + arow;
        const float bias = b1[col];
#pragma unroll
        for (int j = 0; j < 8; ++j) {
            int rowm = (lane < 16) ? j : (j + 8);
            float hv = acc[t][j] + bias;
            float ge = 0.5f * hv * (1.0f + erff(hv * 0.70710678118654752f));
            buf[rowm * PP + col] = ge;
        }
    }
    __syncthreads();

    // ---- GEMM2: (16xP)@(PxP) ----
    v8f acc2[3];
#pragma unroll
    for (int t = 0; t < 3; ++t) {
        const int n0 = (wid + t * 8) * 16;
        v8f c = czero;
        for (int k = 0; k < PP; k += 4) {
            v2f a, bfr;
            a[0]   = buf[arow * PP + k + kh];
            a[1]   = buf[arow * PP + k + kh + 1];
            bfr[0] = W2[(size_t)(k + kh    ) * PP + n0 + arow];
            bfr[1] = W2[(size_t)(k + kh + 1) * PP + n0 + arow];
            c = __builtin_amdgcn_wmma_f32_16x16x4_f32(false, a, false, bfr,
                                                      (short)0, c, false, false);
        }
        acc2[t] = c;
    }
    __syncthreads();

    // ---- y = acc2 + b2 into buf[16*P .. 32*P) ----
#pragma unroll
    for (int t = 0; t < 3; ++t) {
        const int col = (wid + t * 8) * 16 + arow;
        const float bias = b2[col];
#pragma unroll
        for (int j = 0; j < 8; ++j) {
            int rowm = (lane < 16) ? j : (j + 8);
            buf[16 * PP + rowm * PP + col] = acc2[t][j] + bias;
        }
    }
    __syncthreads();

    // ---- L2 normalize rows, write z ----
    for (int r = wid; r < 16; r += 8) {
        float s = 0.0f;
        for (int c = lane; c < PP; c += 32) { float y = buf[16 * PP + r * PP + c]; s += y * y; }
#pragma unroll
        for (int o = 16; o > 0; o >>= 1) s += __shfl_xor(s, o, 32);
        float inv = 1.0f / fmaxf(sqrtf(s), 1e-12f);
        for (int c = lane; c < PP; c += 32)
            zout[r * PP + c] = buf[16 * PP + r * PP + c] * inv;
    }
}

// =====================================================================
// Kernel 4: finalize — CE reduction (fixed order, deterministic),
// InfoNCE on 16x16 sim, BoW BCE, diversity, variance, weighted sum.
// =====================================================================
__global__ __launch_bounds__(256)
void finalize_kernel(const float* __restrict__ tok, const float* __restrict__ vf,
                     const float* __restrict__ z_e, const float* __restrict__ z_t,
                     const float* __restrict__ enc,
                     const int* __restrict__ labels,
                     const float* __restrict__ Wb, const float* __restrict__ bb,
                     float* __restrict__ out)
{
    __shared__ float red[256];
    __shared__ float simm[16][16];
    __shared__ float rowl[16], coll[16], norms[16];
    __shared__ float sh_align;
    const int tid = threadIdx.x;

    // ---- CE ----
    float a = 0.0f, b_ = 0.0f;
    for (int i = tid; i < BB * TT; i += 256) { a += tok[i]; b_ += vf[i]; }
    red[tid] = a; __syncthreads();
    for (int o = 128; o > 0; o >>= 1) { if (tid < o) red[tid] += red[tid + o]; __syncthreads(); }
    float num = red[0]; __syncthreads();
    red[tid] = b_; __syncthreads();
    for (int o = 128; o > 0; o >>= 1) { if (tid < o) red[tid] += red[tid + o]; __syncthreads(); }
    float den = red[0]; __syncthreads();
    float ce = num / fmaxf(den, 1.0f);

    // ---- sim = (z_e @ z_t^T)/tau : thread (i,j) ----
    {
        int i = tid >> 4, j = tid & 15;
        float s = 0.0f;
        for (int k = 0; k < PP; ++k) s += z_e[i * PP + k] * z_t[j * PP + k];
        simm[i][j] = s / TAUV;
    }
    __syncthreads();
    if (tid < 16) {
        float m = -INFINITY;
        for (int j = 0; j < 16; ++j) m = fmaxf(m, simm[tid][j]);
        float s = 0.0f;
        for (int j = 0; j < 16; ++j) s += __expf(simm[tid][j] - m);
        rowl[tid] = m + __logf(s);
    } else if (tid < 32) {
        int j = tid - 16;
        float m = -INFINITY;
        for (int i = 0; i < 16; ++i) m = fmaxf(m, simm[i][j]);
        float s = 0.0f;
        for (int i = 0; i < 16; ++i) s += __expf(simm[i][j] - m);
        coll[j] = m + __logf(s);
    }
    __syncthreads();
    if (tid == 0) {
        float li = 0.0f, lj = 0.0f;
        for (int i = 0; i < 16; ++i) {
            li += -(simm[i][i] - rowl[i]);
            lj += -(simm[i][i] - coll[i]);
        }
        sh_align = 0.5f * (li / 16.0f + lj / 16.0f);
    }
    __syncthreads();

    // ---- BoW BCE: 1024 (b,i) entries ----
    float bces = 0.0f;
    for (int e = tid; e < BB * NBOW; e += 256) {
        int b  = e / NBOW, i = e % NBOW;
        int id = i * 500;
        float tgt = 0.0f;
        for (int t = 0; t < TT; ++t) {
            int lb = labels[b * TT + t];
            bool valid = (lb != 0) && (lb != -100);
            int lc = min(max(lb, 0), VV - 1);
            if (valid && lc == id) { tgt = 1.0f; break; }
        }
        float bl = bb[i];
        for (int k = 0; k < HH; ++k) bl += enc[b * HH + k] * Wb[k * NBOW + i];
        bces += fmaxf(bl, 0.0f) - bl * tgt + log1pf(__expf(-fabsf(bl)));
    }
    red[tid] = bces; __syncthreads();
    for (int o = 128; o > 0; o >>= 1) { if (tid < o) red[tid] += red[tid + o]; __syncthreads(); }
    float bce = red[0] / (float)(BB * NBOW); __syncthreads();

    // ---- diversity: mean |off-diag cos sim| ----
    if (tid < 16) {
        float s = 0.0f;
        for (int k = 0; k < HH; ++k) { float v = enc[tid * HH + k]; s += v * v; }
        norms[tid] = fmaxf(sqrtf(s), 1e-12f);
    }
    __syncthreads();
    float divs = 0.0f;
    {
        int i = tid >> 4, j = tid & 15;
        if (i != j) {
            float s = 0.0f;
            for (int k = 0; k < HH; ++k) s += enc[i * HH + k] * enc[j * HH + k];
            divs = fabsf(s / (norms[i] * norms[j]));
        }
    }
    red[tid] = divs; __syncthreads();
    for (int o = 128; o > 0; o >>= 1) { if (tid < o) red[tid] += red[tid + o]; __syncthreads(); }
    float div = red[0] / (float)(BB * BB - BB); __syncthreads();

    // ---- variance loss: mean exp(-var_dim), ddof=1 ----
    float vs = 0.0f;
    for (int d = tid; d < HH; d += 256) {
        float mean = 0.0f;
        for (int b = 0; b < BB; ++b) mean += enc[b * HH + d];
        mean /= (float)BB;
        float v = 0.0f;
        for (int b = 0; b < BB; ++b) { float dd = enc[b * HH + d] - mean; v += dd * dd; }
        v /= (float)(BB - 1);
        vs += __expf(-v);
    }
    red[tid] = vs; __syncthreads();
    for (int o = 128; o > 0; o >>= 1) { if (tid < o) red[tid] += red[tid + o]; __syncthreads(); }
    float var_l = red[0] / (float)HH;

    if (tid == 0)
        out[0] = W_CE * ce + W_AL * sh_align + W_BOW * bce + W_DIV * div + W_VAR * var_l;
}

// =====================================================================
// launch
// =====================================================================
extern "C" void kernel_launch(void* const* d_in, const int* in_sizes, int n_in,
                              void* d_out, int out_size, void* d_ws, size_t ws_size,
                              hipStream_t stream)
{
    const float* logits = (const float*)d_in[0];
    const int*   labels = (const int*)  d_in[1];
    const int*   amask  = (const int*)  d_in[2];
    const float* enc    = (const float*)d_in[3];
    const float* dh     = (const float*)d_in[4];
    const float* ln_g_e = (const float*)d_in[5];
    const float* ln_b_e = (const float*)d_in[6];
    const float* W1_e   = (const float*)d_in[7];
    const float* b1_e   = (const float*)d_in[8];
    const float* W2_e   = (const float*)d_in[9];
    const float* b2_e   = (const float*)d_in[10];
    const float* ln_g_t = (const float*)d_in[11];
    const float* ln_b_t = (const float*)d_in[12];
    const float* W1_t   = (const float*)d_in[13];
    const float* b1_t   = (const float*)d_in[14];
    const float* W2_t   = (const float*)d_in[15];
    const float* b2_t   = (const float*)d_in[16];
    const float* W_bow  = (const float*)d_in[17];
    const float* b_bow  = (const float*)d_in[18];
    float* out = (float*)d_out;

    float* ws     = (float*)d_ws;
    float* tok    = ws;                 // 2048
    float* vfv    = ws + 2048;          // 2048
    float* pooled = ws + 4096;          // 16*768 = 12288
    float* z_e    = ws + 16384;         // 16*384 = 6144
    float* z_t    = ws + 22528;         // 16*384 = 6144

    ce_kernel<<<BB * TT, 256, 0, stream>>>(logits, labels, tok, vfv);
    pool_kernel<<<BB, 256, 0, stream>>>(dh, amask, pooled);
    proj_wmma_kernel<<<1, 256, 0, stream>>>(enc,    ln_g_e, ln_b_e, W1_e, b1_e, W2_e, b2_e, z_e);
    proj_wmma_kernel<<<1, 256, 0, stream>>>(pooled, ln_g_t, ln_b_t, W1_t, b1_t, W2_t, b2_t, z_t);
    finalize_kernel<<<1, 256, 0, stream>>>(tok, vfv, z_e, z_t, enc, labels, W_bow, b_bow, out);
}